// GCN_6794638262429
// MI455X (gfx1250) — compile-verified
//
#include <hip/hip_runtime.h>

#define IN_DIM 128
#define HID    128
#define OUT_DIM 40

typedef float v2f __attribute__((ext_vector_type(2)));
typedef float v8f __attribute__((ext_vector_type(8)));

// LDS byte offset of a generic pointer that points into shared memory.
__device__ __forceinline__ unsigned lds_byte_off(const void* p) {
  return (unsigned)(uintptr_t)(__attribute__((address_space(3))) const void*)p;
}

// ------------------------------------------------------------------
// Degree / symmetric normalization: deg over dst (+1 self loop),
// dis = deg^{-1/2}, norm(e) = dis[src]*dis[dst].
// ------------------------------------------------------------------
__global__ void k_deg_init(float* __restrict__ deg, int n) {
  int i = blockIdx.x * blockDim.x + threadIdx.x;
  if (i < n) deg[i] = 1.0f;  // self loop contribution
}

__global__ void k_deg_accum(const long long* __restrict__ dst,
                            float* __restrict__ deg, int E) {
  int i = blockIdx.x * blockDim.x + threadIdx.x;
  if (i < E) atomicAdd(&deg[(int)dst[i]], 1.0f);
}

__global__ void k_deg_to_dis(float* __restrict__ deg, int n) {
  int i = blockIdx.x * blockDim.x + threadIdx.x;
  if (i < n) {
    float d = deg[i];
    deg[i] = d > 0.0f ? rsqrtf(d) : 0.0f;
  }
}

// ------------------------------------------------------------------
// WMMA fp32 GEMM: D[n_rows,n_cols] = (relu?)A[n_rows,128] @ W[128,n_cols]
// Block = 256 threads = 8 waves, one 16-row strip per wave.
// W is staged ONCE per block into LDS *in fragment order*: entry
// (kk, t, lane) is the v2f pair {W[4kk+2h][16t+mn], W[4kk+2h+1][16t+mn]}
// (h = lane>>4, mn = lane&15). Each B fragment is then a single aligned
// ds_load_b64 (consecutive lanes -> consecutive 8B -> conflict-free),
// feeding V_WMMA_F32_16X16X4_F32 with no shuffle movs.
// Fragment layouts per CDNA5 ISA 7.12.2:
//   A 16x4 : lane L -> M = L%16, holds K = 2*(L/16)+{0,1}
//   B 4x16 : lane L -> N = L%16, holds K rows 2*(L/16)+{0,1}
//   D 16x16: VGPR v, lane L -> M = v + 8*(L/16), N = L%16
// ------------------------------------------------------------------
template <int NT, bool RELU_IN, bool GUARD>
__global__ __launch_bounds__(256)
void k_gemm_wmma(const float* __restrict__ A,   // [n_rows, 128]
                 const float* __restrict__ W,   // [128, n_cols] row-major
                 const float* __restrict__ bias,// nullptr or [n_cols]
                 float* __restrict__ D,         // [n_rows, n_cols]
                 int n_strips, int n_cols) {
  constexpr int NC   = NT * 16;            // padded column count (128 or 48)
  constexpr int NKK  = IN_DIM / 4;         // 32 K-steps
  constexpr int NENT = NKK * NT * 32;      // fragment entries (8192 / 3072)
  __shared__ v2f lds_b[NENT];              // 64 KB (NT=8) / 24 KB (NT=3)

  const int tid  = threadIdx.x;
  const int lane = tid & 31;
  const int wave = tid >> 5;

  // ---- stage W into LDS in fragment order ----
  if (!GUARD) {
    // n_cols == NC: async global->LDS b32 scatters (ASYNCcnt)
    for (int e = tid; e < NENT; e += 256) {
      const int kk = e / (NT * 32);
      const int r  = e % (NT * 32);
      const int t  = r >> 5;
      const int le = r & 31;
      const int k   = 4 * kk + 2 * (le >> 4);
      const int col = 16 * t + (le & 15);
      const unsigned loff = (unsigned)e * 8u;
      unsigned long long g0 =
          (unsigned long long)(uintptr_t)(W + (size_t)k * NC + col);
      asm volatile("global_load_async_to_lds_b32 %0, %1, off"
                   :: "v"(loff), "v"(g0) : "memory");
      unsigned long long g1 = g0 + (unsigned long long)NC * 4u;
      asm volatile("global_load_async_to_lds_b32 %0, %1, off offset:4"
                   :: "v"(loff), "v"(g1) : "memory");
    }
    asm volatile("s_wait_asynccnt 0x0" ::: "memory");
  } else {
    // ragged n_cols (final 128x40): elementwise with zero padding
    for (int e = tid; e < NENT; e += 256) {
      const int kk = e / (NT * 32);
      const int r  = e % (NT * 32);
      const int t  = r >> 5;
      const int le = r & 31;
      const int k   = 4 * kk + 2 * (le >> 4);
      const int col = 16 * t + (le & 15);
      v2f p;
      p.x = (col < n_cols) ? W[(size_t)k * n_cols + col] : 0.0f;
      p.y = (col < n_cols) ? W[(size_t)(k + 1) * n_cols + col] : 0.0f;
      lds_b[e] = p;
    }
  }
  __syncthreads();

  const int strip = blockIdx.x * 8 + wave;      // one strip per wave
  if (strip >= n_strips) return;                // wave-uniform; EXEC all-1s below
  const int h   = lane >> 4;                    // lane half (0/1)
  const int mn  = lane & 15;                    // M for A-frag, N for B/D-frag
  const int row0 = strip * 16;

  // per-lane A base: row (row0+mn), K offset 2*h; step 4 per K-iteration
  const v2f* __restrict__ ap =
      (const v2f*)(A + (size_t)(row0 + mn) * IN_DIM + 2 * h);
  // per-lane B fragment base: lds_b[lane + (kk*NT + t)*32] (const byte offsets)
  const v2f* __restrict__ bfrag = lds_b + lane;

  const v8f vzero = {0.f, 0.f, 0.f, 0.f, 0.f, 0.f, 0.f, 0.f};
  v8f acc[NT];
#pragma unroll
  for (int t = 0; t < NT; ++t) acc[t] = vzero;

#pragma unroll 8
  for (int kk = 0; kk < NKK; ++kk) {
    v2f a = ap[2 * kk];                         // A[row, 4kk+2h .. +1]
    if (RELU_IN) { a.x = fmaxf(a.x, 0.f); a.y = fmaxf(a.y, 0.f); }
#pragma unroll
    for (int t = 0; t < NT; ++t) {
      v2f b = bfrag[(kk * NT + t) * 32];        // single ds_load_b64
      acc[t] = __builtin_amdgcn_wmma_f32_16x16x4_f32(
          /*neg_a=*/false, a, /*neg_b=*/false, b,
          /*c_mod=*/(short)0, acc[t], /*reuse_a=*/false, /*reuse_b=*/false);
    }
  }

#pragma unroll
  for (int t = 0; t < NT; ++t) {
    const int col = t * 16 + mn;
    if (!GUARD || col < n_cols) {
      float bv = (bias != nullptr) ? bias[col] : 0.0f;
#pragma unroll
      for (int v = 0; v < 8; ++v) {
        D[(size_t)(row0 + 8 * h + v) * n_cols + col] = acc[t][v] + bv;
      }
    }
  }
}

// ------------------------------------------------------------------
// AGG[i,f] = H[i,f] * dis[i]^2 + bias[f]   (self-loop term + bias)
// ------------------------------------------------------------------
__global__ void k_self_bias(const float* __restrict__ H,
                            const float* __restrict__ dis,
                            const float* __restrict__ bias,
                            float* __restrict__ AGG, size_t total) {
  size_t t = (size_t)blockIdx.x * blockDim.x + threadIdx.x;
  if (t < total) {
    int i = (int)(t >> 7);
    int f = (int)(t & 127);
    float s = dis[i];
    AGG[t] = H[t] * s * s + bias[f];
  }
}

// ------------------------------------------------------------------
// Edge scatter: one wave per edge; float4 gather of H[src], scaled
// atomic float adds into AGG[dst] (51 MB working set resolves in L2).
// ------------------------------------------------------------------
__global__ void k_edge_scatter(const long long* __restrict__ src,
                               const long long* __restrict__ dst,
                               const float* __restrict__ dis,
                               const float* __restrict__ H,
                               float* __restrict__ AGG, int E) {
  int t = blockIdx.x * blockDim.x + threadIdx.x;
  int e = t >> 5;       // 32 lanes cover 128 features (float4 each)
  int lane = t & 31;
  if (e >= E) return;
  if (lane == 0 && e + 4096 < E) {   // warm the index stream (global_prefetch_b8)
    __builtin_prefetch(&src[e + 4096], 0, 1);
    __builtin_prefetch(&dst[e + 4096], 0, 1);
  }
  int s = (int)src[e];
  int d = (int)dst[e];
  float w = dis[s] * dis[d];
  const float4* __restrict__ hs = (const float4*)(H + (size_t)s * HID);
  float4 v = hs[lane];
  float* ag = AGG + (size_t)d * HID + lane * 4;
  atomicAdd(ag + 0, v.x * w);
  atomicAdd(ag + 1, v.y * w);
  atomicAdd(ag + 2, v.z * w);
  atomicAdd(ag + 3, v.w * w);
}

// ------------------------------------------------------------------
// Driver: deg -> dis; 3x (GEMM -> self+bias -> edge scatter); final linear.
// ReLU of each layer output is fused into the next GEMM's A-load.
// ------------------------------------------------------------------
extern "C" void kernel_launch(void* const* d_in, const int* in_sizes, int n_in,
                              void* d_out, int out_size, void* d_ws, size_t ws_size,
                              hipStream_t stream) {
  const float*     x  = (const float*)d_in[0];
  const long long* ei = (const long long*)d_in[1];
  const float*     W1 = (const float*)d_in[2];
  const float*     b1 = (const float*)d_in[3];
  const float*     W2 = (const float*)d_in[4];
  const float*     b2 = (const float*)d_in[5];
  const float*     W3 = (const float*)d_in[6];
  const float*     b3 = (const float*)d_in[7];
  const float*     Wl = (const float*)d_in[8];
  const float*     bl = (const float*)d_in[9];

  const int N = in_sizes[0] / IN_DIM;
  const int E = in_sizes[1] / 2;
  const long long* srcp = ei;
  const long long* dstp = ei + E;

  // workspace: dis[N] | H[N,128] | AGG[N,128]  (~103 MB)
  char* ws = (char*)d_ws;
  float* dis = (float*)ws;
  size_t off = (((size_t)N * 4) + 255) & ~(size_t)255;
  float* H = (float*)(ws + off);
  off += (size_t)N * HID * sizeof(float);
  float* AGG = (float*)(ws + off);

  k_deg_init  <<<(N + 255) / 256, 256, 0, stream>>>(dis, N);
  k_deg_accum <<<(E + 255) / 256, 256, 0, stream>>>(dstp, dis, E);
  k_deg_to_dis<<<(N + 255) / 256, 256, 0, stream>>>(dis, N);

  const int strips = N / 16;                    // 100000/16 = 6250
  const int gemm_blocks = (strips + 7) / 8;     // 8 waves (strips) per block
  const size_t totalNH = (size_t)N * HID;
  const int sb_blocks = (int)((totalNH + 255) / 256);
  const long long ethreads = (long long)E * 32;
  const int e_blocks = (int)((ethreads + 255) / 256);

  // layer 1
  k_gemm_wmma<8, false, false><<<gemm_blocks, 256, 0, stream>>>(x,   W1, nullptr, H, strips, HID);
  k_self_bias<<<sb_blocks, 256, 0, stream>>>(H, dis, b1, AGG, totalNH);
  k_edge_scatter<<<e_blocks, 256, 0, stream>>>(srcp, dstp, dis, H, AGG, E);
  // layer 2 (ReLU fused on input)
  k_gemm_wmma<8, true, false><<<gemm_blocks, 256, 0, stream>>>(AGG, W2, nullptr, H, strips, HID);
  k_self_bias<<<sb_blocks, 256, 0, stream>>>(H, dis, b2, AGG, totalNH);
  k_edge_scatter<<<e_blocks, 256, 0, stream>>>(srcp, dstp, dis, H, AGG, E);
  // layer 3 (ReLU fused on input, linear output)
  k_gemm_wmma<8, true, false><<<gemm_blocks, 256, 0, stream>>>(AGG, W3, nullptr, H, strips, HID);
  k_self_bias<<<sb_blocks, 256, 0, stream>>>(H, dis, b3, AGG, totalNH);
  k_edge_scatter<<<e_blocks, 256, 0, stream>>>(srcp, dstp, dis, H, AGG, E);
  // final linear: d_out = AGG @ Wl + bl   (40 cols -> 3 guarded tiles)
  k_gemm_wmma<3, false, true><<<gemm_blocks, 256, 0, stream>>>(AGG, Wl, bl, (float*)d_out, strips, OUT_DIM);
}